// GNNModel_73555609911826
// MI455X (gfx1250) — compile-verified
//
#include <hip/hip_runtime.h>

typedef float v2f __attribute__((ext_vector_type(2)));
typedef float v8f __attribute__((ext_vector_type(8)));

#define N_NODES 100000
#define F_IN 256
#define H1 64
#define H2 32
#define NEG_SLOPE 0.2f

// ---------------------------------------------------------------------------
// float atomic max via integer ordering trick (init value must be -inf)
// ---------------------------------------------------------------------------
__device__ __forceinline__ void atomicMaxF(float* addr, float v) {
  if (v >= 0.0f)
    atomicMax((int*)addr, __float_as_int(v));
  else
    atomicMin((unsigned int*)addr, __float_as_uint(v));
}

// ---------------------------------------------------------------------------
// WMMA f32 GEMM: Hout[M x NOUT] = X[M x K] @ W[K x NOUT]
// Block: (NOUT/16) waves; each wave owns a 16-col stripe and MROWS/16 row
// tiles, sharing one B fragment per k-step across all row tiles (B reuse).
// LDS row stride padded (+4 floats) so the 16 lanes of a half-wave hit
// distinct banks on the A-fragment ds_loads.
// A fragment (16x4 f32): lanes 0-15 hold K=k,k+1 ; lanes 16-31 hold K=k+2,k+3
// B fragment (4x16 f32): VGPR0 = row k (lanes 0-15) / row k+2 (lanes 16-31)
// C/D (16x16 f32, 8 VGPRs): VGPR r -> M = r (lanes 0-15) / r+8 (lanes 16-31)
// ---------------------------------------------------------------------------
template <int K, int NOUT, int MROWS>
__global__ void gemm_wmma_kernel(const float* __restrict__ X,
                                 const float* __restrict__ W,
                                 float* __restrict__ Hout) {
  constexpr int LDK = K + 4;        // padded LDS row stride (bank spread)
  constexpr int RT = MROWS / 16;    // row tiles per wave
  __shared__ float xs[MROWS * LDK];
  const int nthreads = (NOUT / 16) * 32;
  const int tid = threadIdx.x;
  const int row0 = blockIdx.x * MROWS;

  __builtin_prefetch(W, 0, 1);  // global_prefetch_b8 of the L2-resident weights

  // stage the MROWS x K activation tile in LDS (coalesced global reads)
  for (int i = tid; i < MROWS * K; i += nthreads) {
    const int r = i / K, c = i % K;
    xs[r * LDK + c] = X[(size_t)(row0 + r) * K + c];
  }
  __syncthreads();

  const int wave = tid >> 5;
  const int lane = tid & 31;
  const int lh = lane & 15;
  const bool hi = (lane >= 16);
  const int col0 = wave * 16;

  v8f acc[RT];
#pragma unroll
  for (int t = 0; t < RT; ++t) acc[t] = (v8f){};

#pragma unroll 4
  for (int k = 0; k < K; k += 4) {
    const int ka = k + (hi ? 2 : 0);
    v2f b;
    b.x = W[(size_t)ka * NOUT + col0 + lh];
    b.y = W[(size_t)(ka + 1) * NOUT + col0 + lh];
#pragma unroll
    for (int t = 0; t < RT; ++t) {
      v2f a;
      a.x = xs[(t * 16 + lh) * LDK + ka];
      a.y = xs[(t * 16 + lh) * LDK + ka + 1];
      acc[t] = __builtin_amdgcn_wmma_f32_16x16x4_f32(false, a, false, b,
                                                     (short)0, acc[t],
                                                     false, false);
    }
  }

  const int rbase = hi ? 8 : 0;
#pragma unroll
  for (int t = 0; t < RT; ++t) {
#pragma unroll
    for (int r = 0; r < 8; ++r) {
      Hout[(size_t)(row0 + t * 16 + rbase + r) * NOUT + col0 + lh] = acc[t][r];
    }
  }
}

// ---------------------------------------------------------------------------
// per-node attention logits: as[n] = h[n]·a_src , ad[n] = h[n]·a_dst
// ---------------------------------------------------------------------------
template <int NOUT>
__global__ void alpha_kernel(const float* __restrict__ H,
                             const float* __restrict__ av,
                             const float* __restrict__ bv,
                             float* __restrict__ as_, float* __restrict__ ad_,
                             int N) {
  const int n = blockIdx.x * blockDim.x + threadIdx.x;
  if (n >= N) return;
  float s = 0.0f, d = 0.0f;
#pragma unroll
  for (int c = 0; c < NOUT; ++c) {
    const float h = H[(size_t)n * NOUT + c];
    s += h * av[c];
    d += h * bv[c];
  }
  as_[n] = s;
  ad_[n] = d;
}

// ---------------------------------------------------------------------------
// init: m = -inf, denom = 0, agg = 0
// ---------------------------------------------------------------------------
__global__ void init_kernel(float* __restrict__ m, float* __restrict__ den,
                            float* __restrict__ agg, int N, int total) {
  const int i = blockIdx.x * blockDim.x + threadIdx.x;
  if (i < total) agg[i] = 0.0f;
  if (i < N) {
    m[i] = -__builtin_inff();
    den[i] = 0.0f;
  }
}

// ---------------------------------------------------------------------------
// pass 1: e = leaky_relu(as[src] + ad[dst]) ; segment max into m[dst]
// edges [0,E) from edge_index, [E, E+N) are self loops
// ---------------------------------------------------------------------------
__global__ void edge_max_kernel(const int* __restrict__ src,
                                const int* __restrict__ dst, int E, int N,
                                const float* __restrict__ as_,
                                const float* __restrict__ ad_,
                                float* __restrict__ ev, float* __restrict__ m) {
  const int e = blockIdx.x * blockDim.x + threadIdx.x;
  if (e >= E + N) return;
  const int s = (e < E) ? src[e] : (e - E);
  const int d = (e < E) ? dst[e] : (e - E);
  float v = as_[s] + ad_[d];
  v = (v > 0.0f) ? v : NEG_SLOPE * v;
  ev[e] = v;
  atomicMaxF(&m[d], v);
}

// ---------------------------------------------------------------------------
// pass 2: ex = exp(e - m[dst]) ; segment sum into denom[dst]
// ---------------------------------------------------------------------------
__global__ void edge_exp_kernel(const int* __restrict__ dst, int E, int N,
                                const float* __restrict__ ev,
                                const float* __restrict__ m,
                                float* __restrict__ ex,
                                float* __restrict__ den) {
  const int e = blockIdx.x * blockDim.x + threadIdx.x;
  if (e >= E + N) return;
  const int d = (e < E) ? dst[e] : (e - E);
  const float t = __expf(ev[e] - m[d]);
  ex[e] = t;
  atomicAdd(&den[d], t);
}

// ---------------------------------------------------------------------------
// pass 3: agg[dst] += h[src] * (ex / denom[dst]); one thread per (edge, chan)
// channel-major across lanes -> coalesced H reads, L2-resident f32 atomics
// ---------------------------------------------------------------------------
template <int NOUT>
__global__ void edge_agg_kernel(const int* __restrict__ src,
                                const int* __restrict__ dst, int E, int N,
                                const float* __restrict__ H,
                                const float* __restrict__ ex,
                                const float* __restrict__ den,
                                float* __restrict__ agg) {
  const long long i = (long long)blockIdx.x * blockDim.x + threadIdx.x;
  const int e = (int)(i / NOUT);
  const int c = (int)(i - (long long)e * NOUT);
  if (e >= E + N) return;
  const int s = (e < E) ? src[e] : (e - E);
  const int d = (e < E) ? dst[e] : (e - E);
  const float attn = ex[e] / den[d];
  atomicAdd(&agg[(size_t)d * NOUT + c], H[(size_t)s * NOUT + c] * attn);
}

// ---------------------------------------------------------------------------
// out = relu(agg + b)
// ---------------------------------------------------------------------------
template <int NOUT>
__global__ void bias_relu_kernel(float* __restrict__ agg,
                                 const float* __restrict__ b, int total) {
  const int i = blockIdx.x * blockDim.x + threadIdx.x;
  if (i >= total) return;
  const float v = agg[i] + b[i % NOUT];
  agg[i] = (v > 0.0f) ? v : 0.0f;
}

// ---------------------------------------------------------------------------
// final fc: out[n] = h2[n]·Wfc + bfc
// ---------------------------------------------------------------------------
__global__ void fc_kernel(const float* __restrict__ H,
                          const float* __restrict__ Wfc,
                          const float* __restrict__ bfc,
                          float* __restrict__ out, int N) {
  const int n = blockIdx.x * blockDim.x + threadIdx.x;
  if (n >= N) return;
  float sum = bfc[0];
#pragma unroll
  for (int c = 0; c < H2; ++c) sum += H[(size_t)n * H2 + c] * Wfc[c];
  out[n] = sum;
}

// ---------------------------------------------------------------------------
static inline unsigned cdiv(long long a, long long b) {
  return (unsigned)((a + b - 1) / b);
}

extern "C" void kernel_launch(void* const* d_in, const int* in_sizes, int n_in,
                              void* d_out, int out_size, void* d_ws,
                              size_t ws_size, hipStream_t stream) {
  (void)n_in; (void)out_size; (void)ws_size;
  const float* x     = (const float*)d_in[0];
  const int*   ei    = (const int*)d_in[1];
  const float* W1    = (const float*)d_in[2];
  const float* asrc1 = (const float*)d_in[3];
  const float* adst1 = (const float*)d_in[4];
  const float* b1    = (const float*)d_in[5];
  const float* W2    = (const float*)d_in[6];
  const float* asrc2 = (const float*)d_in[7];
  const float* adst2 = (const float*)d_in[8];
  const float* b2    = (const float*)d_in[9];
  const float* Wfc   = (const float*)d_in[10];
  const float* bfc   = (const float*)d_in[11];
  float* out = (float*)d_out;

  const int N = N_NODES;
  const int E = in_sizes[1] / 2;     // edge_index is [2, E]
  const int Etot = E + N;            // + self loops
  const int* src = ei;
  const int* dst = ei + E;

  // workspace layout (floats)
  float* ws = (float*)d_ws;
  size_t off = 0;
  float* h    = ws + off; off += (size_t)N * H1;   // h1, reused for h2
  float* agg1 = ws + off; off += (size_t)N * H1;   // layer-1 output / layer-2 input
  float* agg2 = ws + off; off += (size_t)N * H2;   // layer-2 output
  float* as_  = ws + off; off += N;
  float* ad_  = ws + off; off += N;
  float* m    = ws + off; off += N;
  float* den  = ws + off; off += N;
  float* ev   = ws + off; off += Etot;
  float* ex   = ws + off; off += Etot;

  const dim3 B(256);
  constexpr int MR = 32;             // 100000 = 32 * 3125, no tail

  // ---------------- layer 1 (256 -> 64) ----------------
  gemm_wmma_kernel<F_IN, H1, MR><<<N / MR, (H1 / 16) * 32, 0, stream>>>(x, W1, h);
  alpha_kernel<H1><<<cdiv(N, 256), B, 0, stream>>>(h, asrc1, adst1, as_, ad_, N);
  init_kernel<<<cdiv((long long)N * H1, 256), B, 0, stream>>>(m, den, agg1, N, N * H1);
  edge_max_kernel<<<cdiv(Etot, 256), B, 0, stream>>>(src, dst, E, N, as_, ad_, ev, m);
  edge_exp_kernel<<<cdiv(Etot, 256), B, 0, stream>>>(dst, E, N, ev, m, ex, den);
  edge_agg_kernel<H1><<<cdiv((long long)Etot * H1, 256), B, 0, stream>>>(
      src, dst, E, N, h, ex, den, agg1);
  bias_relu_kernel<H1><<<cdiv((long long)N * H1, 256), B, 0, stream>>>(agg1, b1, N * H1);

  // ---------------- layer 2 (64 -> 32) ----------------
  gemm_wmma_kernel<H1, H2, MR><<<N / MR, (H2 / 16) * 32, 0, stream>>>(agg1, W2, h);
  alpha_kernel<H2><<<cdiv(N, 256), B, 0, stream>>>(h, asrc2, adst2, as_, ad_, N);
  init_kernel<<<cdiv((long long)N * H2, 256), B, 0, stream>>>(m, den, agg2, N, N * H2);
  edge_max_kernel<<<cdiv(Etot, 256), B, 0, stream>>>(src, dst, E, N, as_, ad_, ev, m);
  edge_exp_kernel<<<cdiv(Etot, 256), B, 0, stream>>>(dst, E, N, ev, m, ex, den);
  edge_agg_kernel<H2><<<cdiv((long long)Etot * H2, 256), B, 0, stream>>>(
      src, dst, E, N, h, ex, den, agg2);
  bias_relu_kernel<H2><<<cdiv((long long)N * H2, 256), B, 0, stream>>>(agg2, b2, N * H2);

  // ---------------- fc (32 -> 1) ----------------
  fc_kernel<<<cdiv(N, 256), B, 0, stream>>>(agg2, Wfc, bfc, out, N);
}